// Efron_loss_penalty_no_exp_89833535963288
// MI455X (gfx1250) — compile-verified
//
#include <hip/hip_runtime.h>

// ---------------------------------------------------------------------------
// Efron-style loss (penalty, no exp) for MI455X / gfx1250.
//
// Strategy (memory-bound: 48 MiB stream -> ~2.2 us floor at 23.3 TB/s):
//   k_init  : zero the accumulators in workspace.
//   k_hist  : per-block LDS-privatized 2048-bin histograms (32 KB LDS/WGP of
//             320 KB) fed by b128 vector loads + global_prefetch_b8; flush
//             with one global atomic per non-empty bin.
//   k_group : one block per time group; each thread sums logf(S - (j/d)*R)
//             over its strided j's; wave reduction via V_WMMA_F32_16X16X4_F32
//             (A = ones, B = (partial, 0) => D[m][n] = colsum_n) + 4 shfl_xor;
//             group term accumulated into an f64 via global_atomic_add_f64.
//   k_final : write [loss, d(2048), S(2048), arange(2048)] as floats.
// ---------------------------------------------------------------------------

#define TMAXC 2048
#define PENALTY_F 0.01f

typedef __attribute__((ext_vector_type(2))) float v2f;
typedef __attribute__((ext_vector_type(8))) float v8f;

struct Ws {
    float  S[TMAXC];
    float  R[TMAXC];
    float  A[TMAXC];
    int    d[TMAXC];
    double loss;      // byte offset 32768, 8-aligned
};

__global__ void k_init(Ws* __restrict__ ws) {
    int i = blockIdx.x * blockDim.x + threadIdx.x;
    if (i < TMAXC) {
        ws->S[i] = 0.0f;
        ws->R[i] = 0.0f;
        ws->A[i] = 0.0f;
        ws->d[i] = 0;
    }
    if (i == 0) ws->loss = 0.0;
}

__device__ __forceinline__ void bin_one(float* sS, float* sR, float* sA, int* sd,
                                        int t, int e, float r) {
    t &= (TMAXC - 1);                 // inputs are in [0, 2048); mask for safety
    atomicAdd(&sS[t], r);
    atomicAdd(&sA[t], fabsf(r));
    if (e) {
        atomicAdd(&sR[t], r);
        atomicAdd(&sd[t], 1);
    }
}

__global__ void k_hist(const int* __restrict__ times,
                       const int* __restrict__ events,
                       const float* __restrict__ risk,
                       int n, Ws* __restrict__ ws) {
    __shared__ float sS[TMAXC];
    __shared__ float sR[TMAXC];
    __shared__ float sA[TMAXC];
    __shared__ int   sd[TMAXC];

    for (int b = threadIdx.x; b < TMAXC; b += blockDim.x) {
        sS[b] = 0.0f; sR[b] = 0.0f; sA[b] = 0.0f; sd[b] = 0;
    }
    __syncthreads();

    const int gtid   = blockIdx.x * blockDim.x + threadIdx.x;
    const int stride = gridDim.x * blockDim.x;
    const int nvec   = n >> 2;

    const int4*   t4 = (const int4*)times;
    const int4*   e4 = (const int4*)events;
    const float4* r4 = (const float4*)risk;

    for (int i = gtid; i < nvec; i += stride) {
        if (i + stride < nvec) {
            // gfx1250 global_prefetch_b8: stay ahead of the 48 MiB stream
            __builtin_prefetch(&t4[i + stride], 0, 3);
            __builtin_prefetch(&e4[i + stride], 0, 3);
            __builtin_prefetch(&r4[i + stride], 0, 3);
        }
        int4   t = t4[i];
        int4   e = e4[i];
        float4 r = r4[i];
        bin_one(sS, sR, sA, sd, t.x, e.x, r.x);
        bin_one(sS, sR, sA, sd, t.y, e.y, r.y);
        bin_one(sS, sR, sA, sd, t.z, e.z, r.z);
        bin_one(sS, sR, sA, sd, t.w, e.w, r.w);
    }
    // scalar tail (n not multiple of 4)
    for (int i = (nvec << 2) + gtid; i < n; i += stride) {
        bin_one(sS, sR, sA, sd, times[i], events[i], risk[i]);
    }

    __syncthreads();

    for (int b = threadIdx.x; b < TMAXC; b += blockDim.x) {
        float s = sS[b]; if (s != 0.0f) atomicAdd(&ws->S[b], s);
        float r = sR[b]; if (r != 0.0f) atomicAdd(&ws->R[b], r);
        float a = sA[b]; if (a != 0.0f) atomicAdd(&ws->A[b], a);
        int   c = sd[b]; if (c != 0)    atomicAdd(&ws->d[b], c);
    }
}

// One block (256 threads = 8 wave32) per time group.
__global__ void k_group(Ws* __restrict__ ws) {
    const int g = blockIdx.x;

    const float Sg = ws->S[g];
    const float Rg = ws->R[g];
    const float Ag = ws->A[g];
    const int   dg = ws->d[g];

    const float df = (float)(dg > 0 ? dg : 1);

    float part = 0.0f;
    for (int j = threadIdx.x; j < dg; j += blockDim.x) {
        float denom = Sg - ((float)j / df) * Rg;   // matches reference (j/d_f)*R
        part += logf(denom);
    }

    // ---- wave-level reduction via fp32 WMMA ----
    // A = ones (16x4), B = (part in VGPR0, 0 in VGPR1) => D[m][n] = colsum_n =
    // part(lane n) + part(lane n+16); every lane's c[0] holds colsum_{lane&15}.
    v2f a; a.x = 1.0f; a.y = 1.0f;
    v2f b; b.x = part; b.y = 0.0f;
    v8f c = {};
    c = __builtin_amdgcn_wmma_f32_16x16x4_f32(
            /*neg_a=*/false, a, /*neg_b=*/false, b,
            /*c_mod=*/(short)0, c, /*reuse_a=*/false, /*reuse_b=*/false);

    float v = c[0];
    v += __shfl_xor(v, 8, 32);
    v += __shfl_xor(v, 4, 32);
    v += __shfl_xor(v, 2, 32);
    v += __shfl_xor(v, 1, 32);   // v = wave total (in every lane)

    __shared__ float sW[8];
    const int lane = threadIdx.x & 31;
    const int wave = threadIdx.x >> 5;
    if (lane == 0) sW[wave] = v;
    __syncthreads();

    if (threadIdx.x == 0) {
        float sum_logs = 0.0f;
        const int nw = blockDim.x >> 5;
        for (int w = 0; w < nw; ++w) sum_logs += sW[w];
        if (dg > 0) {
            float term = PENALTY_F * Ag - Rg + sum_logs;
            atomicAdd(&ws->loss, (double)term);   // global_atomic_add_f64
        }
    }
}

__global__ void k_final(const Ws* __restrict__ ws, float* __restrict__ out) {
    int g = blockIdx.x * blockDim.x + threadIdx.x;
    if (g == 0) out[0] = (float)ws->loss;
    if (g < TMAXC) {
        out[1 + g]             = (float)ws->d[g];  // d (int -> float out buffer)
        out[1 + TMAXC + g]     = ws->S[g];         // S
        out[1 + 2 * TMAXC + g] = (float)g;         // failure_time = arange
    }
}

extern "C" void kernel_launch(void* const* d_in, const int* in_sizes, int n_in,
                              void* d_out, int out_size, void* d_ws, size_t ws_size,
                              hipStream_t stream) {
    const int*   times  = (const int*)d_in[0];
    const int*   events = (const int*)d_in[1];
    const float* risk   = (const float*)d_in[2];
    float*       out    = (float*)d_out;
    Ws*          ws     = (Ws*)d_ws;
    const int    n      = in_sizes[0];

    k_init<<<(TMAXC + 255) / 256, 256, 0, stream>>>(ws);

    // 1024 blocks x 256 threads (8 wave32 each): 16 vec4 elements per thread,
    // 32 KB LDS per block leaves room for multiple blocks per 320 KB WGP.
    k_hist<<<1024, 256, 0, stream>>>(times, events, risk, n, ws);

    k_group<<<TMAXC, 256, 0, stream>>>(ws);

    k_final<<<(TMAXC + 255) / 256, 256, 0, stream>>>(ws, out);
}